// HeadLoss_68865505624238
// MI455X (gfx1250) — compile-verified
//
#include <hip/hip_runtime.h>
#include <stdint.h>

#define THREADS 256
#define NWAVES  8

typedef float v2f  __attribute__((ext_vector_type(2)));
typedef float v8f  __attribute__((ext_vector_type(8)));
typedef unsigned int u32x4 __attribute__((ext_vector_type(4)));
typedef int   i32x4 __attribute__((ext_vector_type(4)));
typedef int   i32x8 __attribute__((ext_vector_type(8)));

#if __has_builtin(__builtin_amdgcn_tensor_load_to_lds) && __has_builtin(__builtin_amdgcn_s_wait_tensorcnt)
#define HAVE_TDM 1
#endif

// ---------------------------------------------------------------------------
// Exact fp32 32-lane sum through the matrix pipe (V_WMMA_F32_16X16X4_F32).
// A[m][0]=v[m] (lanes 0-15), A[m][2]=v[m+16] (lanes 16-31), other K slots 0.
// B = ones(4x16)  =>  D[m][n] = v[m] + v[m+16] broadcast along n.
// Summing the 8 D VGPRs per lane gives half-sums; one xor-16 shuffle totals.
// ---------------------------------------------------------------------------
__device__ __forceinline__ float wave_sum_wmma(float v) {
#if __has_builtin(__builtin_amdgcn_wmma_f32_16x16x4_f32)
  v2f a; a[0] = v;    a[1] = 0.0f;
  v2f o; o[0] = 1.0f; o[1] = 1.0f;
  v8f c = {0.f, 0.f, 0.f, 0.f, 0.f, 0.f, 0.f, 0.f};
  v8f d = __builtin_amdgcn_wmma_f32_16x16x4_f32(false, a, false, o, (short)0, c,
                                                false, false);
  float s = d[0] + d[1] + d[2] + d[3] + d[4] + d[5] + d[6] + d[7];
  s += __shfl_xor(s, 16, 32);
  return s;
#else
  for (int off = 16; off > 0; off >>= 1) v += __shfl_xor(v, off, 32);
  return v;
#endif
}

// ---------------------------------------------------------------------------
// TDM: stage a contiguous run of dwords from global into LDS (1-D tile D#).
// D# layout per CDNA5 ISA §8 (CDNA5 ISA). Wave-level op, EXEC ignored.
// ---------------------------------------------------------------------------
#ifdef HAVE_TDM
__device__ __forceinline__ void tdm_stage(const void* gptr, void* lptr,
                                          unsigned ndwords) {
  unsigned long long ga = (unsigned long long)(uintptr_t)gptr;
  unsigned lds = (unsigned)(uintptr_t)lptr;   // flat-shared low 32 bits = LDS byte addr
  u32x4 g0;
  g0[0] = 1u;                                        // count=1, user mode, no gather
  g0[1] = lds;                                       // lds_addr
  g0[2] = (unsigned)(ga & 0xFFFFFFFFu);              // global_addr[31:0]
  g0[3] = (unsigned)((ga >> 32) & 0x01FFFFFFu)       // global_addr[56:32]
        | (2u << 30);                                // type = 2 ("image")
  i32x8 g1;
  g1[0] = (int)(2u << 16);                           // wg_mask=0, data_size=4B
  g1[1] = (int)((ndwords & 0xFFFFu) << 16);          // tensor_dim0[15:0]
  g1[2] = (int)((ndwords >> 16) | (1u << 16));       // tensor_dim0[31:16], tensor_dim1=1
  g1[3] = (int)((ndwords & 0xFFFFu) << 16);          // tile_dim0 = ndwords
  g1[4] = 0;                                         // tile_dim1/2 unused
  g1[5] = (int)ndwords;                              // tensor_dim0_stride
  g1[6] = 0;
  g1[7] = 0;
  i32x4 g2; g2[0] = 1; g2[1] = 1; g2[2] = 0; g2[3] = 0;          // dims 2/3 = 1
  i32x4 g3; g3[0] = 0; g3[1] = (int)(1u << 16); g3[2] = 0; g3[3] = 0; // dim4 = 1
#if __has_include(<hip/amd_detail/amd_gfx1250_TDM.h>)
  i32x8 g4 = {0, 0, 0, 0, 0, 0, 0, 0};
  __builtin_amdgcn_tensor_load_to_lds(g0, g1, g2, g3, g4, 0);    // clang-23 6-arg
#else
  __builtin_amdgcn_tensor_load_to_lds(g0, g1, g2, g3, 0);        // ROCm7.2 5-arg
#endif
  __builtin_amdgcn_s_wait_tensorcnt(0);
}
#endif

__device__ __forceinline__ void stage_to_lds(const float* g, float* s, int n,
                                             int tid) {
#ifdef HAVE_TDM
  if (tid < 32) tdm_stage(g, s, (unsigned)n);   // wave 0 issues + waits TENSORcnt
#else
  for (int i = tid; i < n; i += THREADS) s[i] = g[i];
#endif
  __syncthreads();
}

// ---------------------------------------------------------------------------
// Block-wide arg-best (max or min) with jax.lax.top_k tie semantics
// (equal values -> lower index wins). Sentinel: li == INT_MAX.
// ---------------------------------------------------------------------------
__device__ __forceinline__ void block_arg_best(float v, int i, bool maximize,
                                               volatile float* redv,
                                               volatile int* redi, int tid,
                                               float& bv, int& bi) {
  for (int off = 16; off > 0; off >>= 1) {
    float ov = __shfl_xor(v, off, 32);
    int   oi = __shfl_xor(i, off, 32);
    bool take = maximize ? (ov > v || (ov == v && oi < i))
                         : (ov < v || (ov == v && oi < i));
    if (take) { v = ov; i = oi; }
  }
  if ((tid & 31) == 0) { redv[tid >> 5] = v; redi[tid >> 5] = i; }
  __syncthreads();
  if (tid == 0) {
    for (int w = 1; w < NWAVES; ++w) {
      float ov = redv[w]; int oi = redi[w];
      bool take = maximize ? (ov > v || (ov == v && oi < i))
                           : (ov < v || (ov == v && oi < i));
      if (take) { v = ov; i = oi; }
    }
    redv[0] = v; redi[0] = i;
  }
  __syncthreads();
  bv = redv[0]; bi = redi[0];
  __syncthreads();   // protect scratch reuse next iteration
}

// ---------------------------------------------------------------------------
// Phase A: per-prediction max-IoU + argmax against 128 GT HBBs (LDS-staged).
// ---------------------------------------------------------------------------
__global__ void __launch_bounds__(THREADS) iou_match_kernel(
    const float* __restrict__ box_pred,     // [B,N,5]
    const float* __restrict__ reg_targets,  // [B,M,4,2]
    float* __restrict__ g_miou,             // [B,N]
    int*   __restrict__ g_match,            // [B,N]
    int N, int M) {
  __shared__ float  s_verts[128 * 8];
  __shared__ float4 s_tbb[128];
  __shared__ float  s_tarea[128];

  const int tid = threadIdx.x;
  const int b = blockIdx.y;
  const int n = blockIdx.x * THREADS + tid;

  if (n < N) __builtin_prefetch(box_pred + ((size_t)b * N + n) * 5, 0, 0);

  stage_to_lds(reg_targets + (size_t)b * M * 8, s_verts, M * 8, tid);

  if (tid < M) {
    const float* v = s_verts + tid * 8;
    float xmn = fminf(fminf(v[0], v[2]), fminf(v[4], v[6]));
    float xmx = fmaxf(fmaxf(v[0], v[2]), fmaxf(v[4], v[6]));
    float ymn = fminf(fminf(v[1], v[3]), fminf(v[5], v[7]));
    float ymx = fmaxf(fmaxf(v[1], v[3]), fmaxf(v[5], v[7]));
    s_tbb[tid] = make_float4(xmn, ymn, xmx, ymx);
    s_tarea[tid] = (xmx - xmn) * (ymx - ymn);
  }
  __syncthreads();

  if (n >= N) return;

  const float* bp = box_pred + ((size_t)b * N + n) * 5;
  float cx = bp[0], cy = bp[1], w = bp[2], h = bp[3], ang = bp[4];
  float sn, cs;
  sincosf(ang, &sn, &cs);
  float dx = 0.5f * w, dy = 0.5f * h;
  float p = dx * cs, q = dy * sn, r = dx * sn, t = dy * cs;
  float x0 = cx - p + q, x1 = cx + p + q, x2 = cx + p - q, x3 = cx - p - q;
  float y0 = cy - r - t, y1 = cy + r - t, y2 = cy + r + t, y3 = cy - r + t;
  float ax1 = fminf(fminf(x0, x1), fminf(x2, x3));
  float ax2 = fmaxf(fmaxf(x0, x1), fmaxf(x2, x3));
  float ay1 = fminf(fminf(y0, y1), fminf(y2, y3));
  float ay2 = fmaxf(fmaxf(y0, y1), fmaxf(y2, y3));
  float areaA = (ax2 - ax1) * (ay2 - ay1);

  float best = -1.0f;
  int bm = 0;
  for (int m = 0; m < M; ++m) {
    float4 tb = s_tbb[m];
    float lx = fmaxf(ax1, tb.x), ly = fmaxf(ay1, tb.y);
    float rx = fminf(ax2, tb.z), ry = fminf(ay2, tb.w);
    float wx = fmaxf(rx - lx, 0.0f), wy = fmaxf(ry - ly, 0.0f);
    float inter = wx * wy;
    float uni = areaA + s_tarea[m] - inter;
    float iou = inter / fmaxf(uni, 1e-7f);
    if (iou > best) { best = iou; bm = m; }   // strict > == argmax first occurrence
  }
  g_miou[(size_t)b * N + n] = best;
  g_match[(size_t)b * N + n] = bm;
}

// ---------------------------------------------------------------------------
// Phase B: per-image top-k sampling + classification / regression loss.
// ---------------------------------------------------------------------------
__global__ void __launch_bounds__(THREADS) sample_loss_kernel(
    const float* __restrict__ box_pred,     // [B,N,5]
    const float* __restrict__ class_pred,   // [B,N,C]
    const float* __restrict__ reg_targets,  // [B,M,4,2]
    const int*   __restrict__ cls_targets,  // [B,M]
    const int*   __restrict__ p_nsamples,   // scalar
    const float* __restrict__ g_miou,       // [B,N]
    const int*   __restrict__ g_match,      // [B,N]
    float* __restrict__ g_cls_img,          // [B]
    float* __restrict__ g_reg_img,          // [B]
    int N, int M, int C) {
  __shared__ float    s_verts[128 * 8];
  __shared__ float    s_cv2[128 * 5];
  __shared__ unsigned s_selbits[16384 / 32];
  __shared__ int      s_pos_idx[256];
  __shared__ int      s_pos_match[256];
  __shared__ int      s_neg_idx[256];
  __shared__ float    s_redv[NWAVES];
  __shared__ int      s_redi[NWAVES];
  __shared__ float    s_wcls[NWAVES], s_wreg[NWAVES];
  __shared__ int      s_np, s_nn;

  const int tid = threadIdx.x;
  const int lane = tid & 31, wave = tid >> 5;
  const int b = blockIdx.x;
  const size_t base = (size_t)b * N;

  stage_to_lds(reg_targets + (size_t)b * M * 8, s_verts, M * 8, tid);

  if (tid < M) {   // vertices -> cv2 (cx,cy,w,h,angle), same math as reference
    const float* v = s_verts + tid * 8;
    float e1x = v[2] - v[0], e1y = v[3] - v[1];
    float e2x = v[6] - v[0], e2y = v[7] - v[1];
    s_cv2[tid * 5 + 0] = (v[0] + v[2] + v[4] + v[6]) * 0.25f;
    s_cv2[tid * 5 + 1] = (v[1] + v[3] + v[5] + v[7]) * 0.25f;
    s_cv2[tid * 5 + 2] = sqrtf(e1x * e1x + e1y * e1y);
    s_cv2[tid * 5 + 3] = sqrtf(e2x * e2x + e2y * e2y);
    s_cv2[tid * 5 + 4] = atan2f(e1y, e1x);
  }
  for (int i = tid; i < 16384 / 32; i += THREADS) s_selbits[i] = 0u;
  if (tid == 0) { s_np = 0; s_nn = 0; }
  __syncthreads();

  int ns = p_nsamples[0];
  int kpos = ns >> 1;        if (kpos > 256) kpos = 256;
  int kneg = ns - (ns >> 1); if (kneg > 256) kneg = 256;

  // ---- positives: iteratively extract highest max_iou >= 0.5 ----
  for (int it = 0; it < kpos; ++it) {
    float lv = -1.0f; int li = 0x7FFFFFFF;
    for (int i = tid; i < N; i += THREADS) {
      if (s_selbits[i >> 5] & (1u << (i & 31))) continue;
      float v = g_miou[base + i];                 // L2-resident scan
      if (v >= 0.5f && v > lv) { lv = v; li = i; }
    }
    float bv; int bi;
    block_arg_best(lv, li, true, s_redv, s_redi, tid, bv, bi);
    if (bi == 0x7FFFFFFF) break;   // remaining picks are invalid filler (loss 0)
    if (tid == 0) {
      s_selbits[bi >> 5] |= 1u << (bi & 31);
      int np = s_np;
      s_pos_idx[np] = bi;
      s_pos_match[np] = g_match[base + bi];
      s_np = np + 1;
    }
    __syncthreads();
  }

  // ---- negatives: iteratively extract lowest max_iou among < 0.5 ----
  for (int it = 0; it < kneg; ++it) {
    float lv = 3.402823466e38f; int li = 0x7FFFFFFF;
    for (int i = tid; i < N; i += THREADS) {
      if (s_selbits[i >> 5] & (1u << (i & 31))) continue;
      float v = g_miou[base + i];
      if (v < 0.5f && v < lv) { lv = v; li = i; }
    }
    float bv; int bi;
    block_arg_best(lv, li, false, s_redv, s_redi, tid, bv, bi);
    if (bi == 0x7FFFFFFF) break;
    if (tid == 0) {
      s_selbits[bi >> 5] |= 1u << (bi & 31);
      s_neg_idx[s_nn] = bi;
      s_nn = s_nn + 1;
    }
    __syncthreads();
  }
  __syncthreads();

  // ---- loss: one wave per sampled row ----
  const int np = s_np, nn = s_nn, ntot = np + nn;
  float acc_cls = 0.0f, acc_reg = 0.0f;
  for (int row = wave; row < ntot; row += NWAVES) {
    int idx, tgt, mv = 0;
    if (row < np) {
      idx = s_pos_idx[row];
      mv  = s_pos_match[row];
      tgt = cls_targets[b * M + mv];
    } else {
      idx = s_neg_idx[row - np];
      tgt = C - 1;   // background
    }
    const float* lg = class_pred + (base + (size_t)idx) * C;
    float mx = -3.402823466e38f;
    for (int c = lane; c < C; c += 32) mx = fmaxf(mx, lg[c]);
    for (int off = 16; off > 0; off >>= 1) mx = fmaxf(mx, __shfl_xor(mx, off, 32));
    float se = 0.0f;
    for (int c = lane; c < C; c += 32) se += expf(lg[c] - mx);
    for (int off = 16; off > 0; off >>= 1) se += __shfl_xor(se, off, 32);
    float nll = (mx + logf(se)) - lg[tgt];
    acc_cls += nll;                              // identical in all lanes
    if (row < np) {                              // smooth-L1 on 5 coords
      float sl = 0.0f;
      if (lane < 5) {
        float d = fabsf(box_pred[(base + (size_t)idx) * 5 + lane] -
                        s_cv2[mv * 5 + lane]);
        sl = (d < 1.0f) ? 0.5f * d * d : d - 0.5f;
      }
      for (int off = 16; off > 0; off >>= 1) sl += __shfl_xor(sl, off, 32);
      acc_reg += sl;
    }
  }
  if (lane == 0) { s_wcls[wave] = acc_cls; s_wreg[wave] = acc_reg; }
  __syncthreads();

  // ---- cross-wave reduce through the matrix pipe (all waves, EXEC all-1) ----
  float pc = (lane < NWAVES) ? s_wcls[lane] : 0.0f;
  float pr = (lane < NWAVES) ? s_wreg[lane] : 0.0f;
  float tc = wave_sum_wmma(pc);
  float tr = wave_sum_wmma(pr);
  if (tid == 0) { g_cls_img[b] = tc; g_reg_img[b] = tr; }
}

// ---------------------------------------------------------------------------
// Finalize: per-image sums -> (total, cls, reg), again via WMMA reduction.
// ---------------------------------------------------------------------------
__global__ void finalize_kernel(const float* __restrict__ g_cls_img,
                                const float* __restrict__ g_reg_img,
                                float* __restrict__ out, int B) {
  int lane = threadIdx.x & 31;
  float pc = (lane < B) ? g_cls_img[lane] : 0.0f;
  float pr = (lane < B) ? g_reg_img[lane] : 0.0f;
  float tc = wave_sum_wmma(pc) / (float)B;
  float tr = wave_sum_wmma(pr) / (float)B;
  if (threadIdx.x == 0) {
    out[0] = tc + tr;   // total_loss
    out[1] = tc;        // classification_loss
    out[2] = tr;        // regression_loss
  }
}

// ---------------------------------------------------------------------------
extern "C" void kernel_launch(void* const* d_in, const int* in_sizes, int n_in,
                              void* d_out, int out_size, void* d_ws,
                              size_t ws_size, hipStream_t stream) {
  const float* box_pred   = (const float*)d_in[0];
  const float* class_pred = (const float*)d_in[1];
  const float* reg_tgt    = (const float*)d_in[2];
  const int*   cls_tgt    = (const int*)d_in[3];
  const int*   p_ns       = (const int*)d_in[4];

  const int B = 8;
  const int N = in_sizes[0] / (5 * B);     // 16384
  const int M = in_sizes[3] / B;           // 128
  const int C = in_sizes[1] / (B * N);     // 81

  float* g_miou    = (float*)d_ws;
  int*   g_match   = (int*)((char*)d_ws + (size_t)B * N * sizeof(float));
  float* g_cls_img = (float*)((char*)d_ws + (size_t)B * N * 8);
  float* g_reg_img = g_cls_img + B;

  dim3 gridA((N + THREADS - 1) / THREADS, B);
  iou_match_kernel<<<gridA, THREADS, 0, stream>>>(box_pred, reg_tgt, g_miou,
                                                  g_match, N, M);
  sample_loss_kernel<<<B, THREADS, 0, stream>>>(box_pred, class_pred, reg_tgt,
                                                cls_tgt, p_ns, g_miou, g_match,
                                                g_cls_img, g_reg_img, N, M, C);
  finalize_kernel<<<1, 32, 0, stream>>>(g_cls_img, g_reg_img, (float*)d_out, B);
}